// CNN4CH_24481313587400
// MI455X (gfx1250) — compile-verified
//
#include <hip/hip_runtime.h>
#include <math.h>

typedef __attribute__((ext_vector_type(2))) float v2f;
typedef __attribute__((ext_vector_type(8))) float v8f;

// ---------------------------------------------------------------------------
// Implicit-GEMM 3x3 stride-2 pad-1 conv + bias + ReLU, NCHW, fp32 WMMA.
// Register-blocked: each wave computes (CPW*16 couts) x (64 positions) with
// 4*CPW V_WMMA_F32_16X16X4_F32 per K-step of 4. Weights staged in LDS in
// A-fragment order, K-chunked (KC) to stay under 64KB static LDS.
// Edge (pad) handling: wave-uniform fast path with no predication when no
// lane touches oh==0 / ow==0; otherwise cndmask-clamped unconditional loads.
// POOL=true: instead of storing the activation map, reduce it (bias+ReLU)
// across positions via shfl_xor + atomicAdd into pooled[b][cout] sums,
// fusing the global average pool into the last conv (saves h3 write+read).
// ---------------------------------------------------------------------------
template<int Cin, int Hin, int Win, int Cout, int Hout, int Wout, int CPW, int KC,
         bool POOL>
__global__ __launch_bounds__(256) void conv3x3s2_wmma(
    const float* __restrict__ in, const float* __restrict__ w,
    const float* __restrict__ bias, float* __restrict__ out)
{
    constexpr int K      = Cin * 9;
    constexpr int CHUNKS = K / KC;
    constexpr int KCB    = KC / 4;        // wmma K-steps per chunk
    constexpr int HWin   = Hin * Win;
    constexpr int HWout  = Hout * Wout;
    constexpr int CG     = Cout / (16 * CPW);
    static_assert(K % KC == 0 && KC % 4 == 0, "bad K chunking");

    __shared__ alignas(16) float wfrag[CPW * KCB * 64]; // [c][kb][lane][2]
    __shared__ alignas(16) int   kofft[K];              // per-k input offset
    __shared__ alignas(16) int   kflag[K];              // bit0: kh==0, bit1: kw==0

    const int tid      = threadIdx.x;
    const int cg       = blockIdx.x % CG;
    const int pchunk   = blockIdx.x / CG;
    const int coutBase = cg * (16 * CPW);

    for (int k = tid; k < K; k += 256) {
        const int cin = k / 9, r9 = k % 9, kh = r9 / 3, kw = r9 % 3;
        kofft[k] = cin * HWin + kh * Win + kw;
        kflag[k] = (kh == 0 ? 1 : 0) | (kw == 0 ? 2 : 0);
    }

    const int lane = tid & 31;
    const int wave = tid >> 5;
    const int n    = lane & 15;
    const int hlf  = lane >> 4;
    const int posBase = (pchunk * 8 + wave) * 64;   // 4 pos-tiles of 16 per wave

    int inB[4], outB[4], bIdx[4], em[4];
    bool edge = false;
    #pragma unroll
    for (int t = 0; t < 4; ++t) {
        const int pos = posBase + t * 16 + n;       // flattened b*Hout*Wout
        const int b   = pos / HWout;
        const int rem = pos - b * HWout;
        const int oh  = rem / Wout;
        const int ow  = rem - oh * Wout;
        inB[t]  = b * (Cin * HWin) + (2 * oh - 1) * Win + (2 * ow - 1);
        outB[t] = b * (Cout * HWout) + rem;
        bIdx[t] = b;                                // uniform within a 16-tile
        em[t]   = (oh == 0 ? 1 : 0) | (ow == 0 ? 2 : 0);
        edge = edge || (em[t] != 0);
    }
    const bool anyEdge = __any((int)edge);

    const v8f vzero = {0.f, 0.f, 0.f, 0.f, 0.f, 0.f, 0.f, 0.f};
    v8f acc[CPW][4];
    #pragma unroll
    for (int c = 0; c < CPW; ++c)
        #pragma unroll
        for (int t = 0; t < 4; ++t) acc[c][t] = vzero;

    for (int ch = 0; ch < CHUNKS; ++ch) {
        __syncthreads();   // previous chunk fully consumed (and kofft visible)
        for (int s = tid; s < CPW * KCB * 64; s += 256) {
            const int c  = s / (KCB * 64);
            const int r0 = s - c * (KCB * 64);
            const int kb = r0 >> 6;
            const int r  = r0 & 63;
            const int l  = r >> 1;
            const int j  = r & 1;
            const int k  = ch * KC + 4 * kb + 2 * (l >> 4) + j;
            wfrag[s] = w[(coutBase + c * 16 + (l & 15)) * K + k];
        }
        __syncthreads();

        if (!anyEdge) {
            #pragma unroll 2
            for (int kb = 0; kb < KCB; ++kb) {
                const int kk   = ch * KC + 4 * kb + 2 * hlf;
                const int off0 = kofft[kk];
                const int off1 = kofft[kk + 1];
                v2f afr[CPW];
                #pragma unroll
                for (int c = 0; c < CPW; ++c)
                    afr[c] = *(const v2f*)&wfrag[(c * KCB + kb) * 64 + lane * 2];
                #pragma unroll
                for (int t = 0; t < 4; ++t) {
                    v2f bfr;
                    bfr.x = in[inB[t] + off0];
                    bfr.y = in[inB[t] + off1];
                    #pragma unroll
                    for (int c = 0; c < CPW; ++c)
                        acc[c][t] = __builtin_amdgcn_wmma_f32_16x16x4_f32(
                            false, afr[c], false, bfr, (short)0, acc[c][t], false, false);
                }
            }
        } else {
            #pragma unroll 2
            for (int kb = 0; kb < KCB; ++kb) {
                const int kk   = ch * KC + 4 * kb + 2 * hlf;
                const int off0 = kofft[kk];
                const int off1 = kofft[kk + 1];
                const int f0   = kflag[kk];
                const int f1   = kflag[kk + 1];
                v2f afr[CPW];
                #pragma unroll
                for (int c = 0; c < CPW; ++c)
                    afr[c] = *(const v2f*)&wfrag[(c * KCB + kb) * 64 + lane * 2];
                #pragma unroll
                for (int t = 0; t < 4; ++t) {
                    const bool i0 = (f0 & em[t]) != 0;
                    const bool i1 = (f1 & em[t]) != 0;
                    const int  a0 = i0 ? 0 : inB[t] + off0;   // clamp, no branch
                    const int  a1 = i1 ? 0 : inB[t] + off1;
                    const float v0 = in[a0];
                    const float v1 = in[a1];
                    v2f bfr;
                    bfr.x = i0 ? 0.f : v0;
                    bfr.y = i1 ? 0.f : v1;
                    #pragma unroll
                    for (int c = 0; c < CPW; ++c)
                        acc[c][t] = __builtin_amdgcn_wmma_f32_16x16x4_f32(
                            false, afr[c], false, bfr, (short)0, acc[c][t], false, false);
                }
            }
        }
    }

    // D layout: VGPR r, lane l -> M = r + 8*(l>>4), N = l&15.
    if constexpr (POOL) {
        // Fused global-average-pool: per (c,r,t), sum relu(acc+bias) over the
        // 16 positions held by this half-wave, then atomically accumulate
        // into out[b*Cout + cout] (mean scale applied by the consumer).
        #pragma unroll
        for (int c = 0; c < CPW; ++c)
            #pragma unroll
            for (int r = 0; r < 8; ++r) {
                const int cout = coutBase + c * 16 + r + 8 * hlf;
                const float bv = bias[cout];
                #pragma unroll
                for (int t = 0; t < 4; ++t) {
                    float v = acc[c][t][r] + bv;
                    v = v > 0.f ? v : 0.f;
                    v += __shfl_xor(v, 1);   // xor masks 1..8 stay within each
                    v += __shfl_xor(v, 2);   // 16-lane half of the wave
                    v += __shfl_xor(v, 4);
                    v += __shfl_xor(v, 8);
                    if (n == 0)
                        atomicAdd(&out[bIdx[t] * Cout + cout], v);
                }
            }
    } else {
        #pragma unroll
        for (int c = 0; c < CPW; ++c)
            #pragma unroll
            for (int r = 0; r < 8; ++r) {
                const int cout = coutBase + c * 16 + r + 8 * hlf;
                const float bv = bias[cout];
                #pragma unroll
                for (int t = 0; t < 4; ++t) {
                    float v = acc[c][t][r] + bv;
                    v = v > 0.f ? v : 0.f;
                    out[outB[t] + cout * HWout] = v;
                }
            }
    }
}

// ---------------------------------------------------------------------------
__global__ __launch_bounds__(256) void zero_kernel(float* __restrict__ p, int nElems)
{
    const int i = blockIdx.x * 256 + threadIdx.x;
    if (i < nElems) p[i] = 0.f;
}

// ---------------------------------------------------------------------------
// FC128->256 (ReLU) then FC256->9, one block. Tiny (~2 MFLOP).
// pooled holds position sums; MEAN_SCALE = 1/(Hout*Wout) applied on load.
// ---------------------------------------------------------------------------
__global__ __launch_bounds__(256) void fc_kernel(
    const float* __restrict__ pooled,
    const float* __restrict__ wl1, const float* __restrict__ bl1,
    const float* __restrict__ wl2, const float* __restrict__ bl2,
    float* __restrict__ rout)
{
    constexpr float MEAN_SCALE = 1.0f / (44.0f * 152.0f);
    __shared__ float pin[32 * 128];
    __shared__ float h1[32 * 256];
    const int tid = threadIdx.x;
    for (int i = tid; i < 32 * 128; i += 256) pin[i] = pooled[i] * MEAN_SCALE;
    __syncthreads();

    const float* wrow = wl1 + tid * 128;   // wl1: [256][128]
    for (int b = 0; b < 32; ++b) {
        float s = bl1[tid];
        for (int k = 0; k < 128; ++k) s += pin[b * 128 + k] * wrow[k];
        h1[b * 256 + tid] = s > 0.f ? s : 0.f;
    }
    __syncthreads();

    for (int idx = tid; idx < 32 * 9; idx += 256) {
        const int b = idx / 9, o = idx % 9;
        float s = bl2[o];
        const float* w2 = wl2 + o * 256;   // wl2: [9][256]
        for (int k = 0; k < 256; ++k) s += h1[b * 256 + k] * w2[k];
        rout[idx] = s;
    }
}

// ---------------------------------------------------------------------------
// Per-matrix 3x3 projection onto SO(3):
// Jacobi eigendecomposition of S = A^T A -> V; U = normalize(A V);
// out = U * diag(1,1,sign(det A)) * V^T  (== reference SVD + last-col flip).
// ---------------------------------------------------------------------------
__global__ void svd_so3_kernel(const float* __restrict__ rin, float* __restrict__ out)
{
    const int b = threadIdx.x;
    if (b >= 32) return;
    float A[9];
    #pragma unroll
    for (int i = 0; i < 9; ++i) A[i] = rin[b * 9 + i];

    float S[9], V[9] = {1,0,0, 0,1,0, 0,0,1};
    #pragma unroll
    for (int i = 0; i < 3; ++i)
        #pragma unroll
        for (int j = 0; j < 3; ++j) {
            float s = 0.f;
            #pragma unroll
            for (int k = 0; k < 3; ++k) s += A[k*3 + i] * A[k*3 + j];
            S[i*3 + j] = s;
        }

    for (int sweep = 0; sweep < 14; ++sweep) {
        #pragma unroll
        for (int m = 0; m < 3; ++m) {
            const int p = (m == 2) ? 1 : 0;
            const int q = (m == 0) ? 1 : 2;
            const float apq = S[p*3 + q];
            if (fabsf(apq) < 1e-30f) continue;
            const float tau = (S[q*3 + q] - S[p*3 + p]) / (2.0f * apq);
            const float t   = ((tau >= 0.f) ? 1.f : -1.f) /
                              (fabsf(tau) + sqrtf(1.f + tau * tau));
            const float c   = 1.0f / sqrtf(1.f + t * t);
            const float s   = t * c;
            const float Spp = S[p*3 + p], Sqq = S[q*3 + q];
            S[p*3 + p] = Spp - t * apq;
            S[q*3 + q] = Sqq + t * apq;
            S[p*3 + q] = S[q*3 + p] = 0.f;
            const int r = 3 - p - q;
            const float Srp = S[r*3 + p], Srq = S[r*3 + q];
            S[r*3 + p] = S[p*3 + r] = c * Srp - s * Srq;
            S[r*3 + q] = S[q*3 + r] = s * Srp + c * Srq;
            #pragma unroll
            for (int i = 0; i < 3; ++i) {
                const float Vip = V[i*3 + p], Viq = V[i*3 + q];
                V[i*3 + p] = c * Vip - s * Viq;
                V[i*3 + q] = s * Vip + c * Viq;
            }
        }
    }

    float lam[3] = {S[0], S[4], S[8]};
    int idx[3] = {0, 1, 2};
    for (int i = 0; i < 2; ++i)
        for (int j = i + 1; j < 3; ++j)
            if (lam[idx[j]] > lam[idx[i]]) { int t = idx[i]; idx[i] = idx[j]; idx[j] = t; }
    float Vs[9];
    for (int c0 = 0; c0 < 3; ++c0)
        for (int i = 0; i < 3; ++i) Vs[i*3 + c0] = V[i*3 + idx[c0]];

    float U[9];
    float nrm2 = 0.f;
    for (int c0 = 0; c0 < 3; ++c0) {
        float col[3];
        for (int i = 0; i < 3; ++i)
            col[i] = A[i*3+0]*Vs[0*3+c0] + A[i*3+1]*Vs[1*3+c0] + A[i*3+2]*Vs[2*3+c0];
        const float nn = sqrtf(col[0]*col[0] + col[1]*col[1] + col[2]*col[2]);
        if (c0 == 2) nrm2 = nn;
        const float inv = (nn > 1e-20f) ? 1.f / nn : 0.f;
        U[0*3 + c0] = col[0] * inv;
        U[1*3 + c0] = col[1] * inv;
        U[2*3 + c0] = col[2] * inv;
    }
    if (nrm2 <= 1e-12f) {   // rank-deficient: complete with u0 x u1
        const float cx = U[3]*U[7] - U[6]*U[4];
        const float cy = U[6]*U[1] - U[0]*U[7];
        const float cz = U[0]*U[4] - U[3]*U[1];
        const float cn = sqrtf(cx*cx + cy*cy + cz*cz);
        const float inv = (cn > 1e-20f) ? 1.f / cn : 0.f;
        U[2] = cx * inv; U[5] = cy * inv; U[8] = cz * inv;
    }

    const float det = A[0]*(A[4]*A[8] - A[5]*A[7])
                    - A[1]*(A[3]*A[8] - A[5]*A[6])
                    + A[2]*(A[3]*A[7] - A[4]*A[6]);
    const float sg = (det >= 0.f) ? 1.f : -1.f;

    for (int i = 0; i < 3; ++i)
        for (int j = 0; j < 3; ++j)
            out[b*9 + i*3 + j] = U[i*3+0]*Vs[j*3+0] + U[i*3+1]*Vs[j*3+1]
                               + sg * U[i*3+2]*Vs[j*3+2];
}

// ---------------------------------------------------------------------------
extern "C" void kernel_launch(void* const* d_in, const int* in_sizes, int n_in,
                              void* d_out, int out_size, void* d_ws, size_t ws_size,
                              hipStream_t stream) {
    const float* x   = (const float*)d_in[0];
    const float* wc1 = (const float*)d_in[1];
    const float* bc1 = (const float*)d_in[2];
    const float* wc2 = (const float*)d_in[3];
    const float* bc2 = (const float*)d_in[4];
    const float* wc3 = (const float*)d_in[5];
    const float* bc3 = (const float*)d_in[6];
    const float* wl1 = (const float*)d_in[7];
    const float* bl1 = (const float*)d_in[8];
    const float* wl2 = (const float*)d_in[9];
    const float* bl2 = (const float*)d_in[10];
    float* outp = (float*)d_out;

    // Workspace layout (bytes).
    char* ws = (char*)d_ws;
    constexpr size_t h1Bytes = (size_t)32 * 32 * 176 * 608 * 4;  // 438,304,768
    constexpr size_t h2Bytes = (size_t)32 * 64 * 88 * 304 * 4;   // 219,152,384
    float* h1     = (float*)ws;
    float* h2     = (float*)(ws + h1Bytes);
    float* pooled = (float*)(ws + h1Bytes + h2Bytes);   // 32*128 position sums
    float* rbuf   = pooled + 32 * 128;                  // 32*9 pre-SVD matrices

    // blocks = (Cout/(16*CPW)) * (B*Hout*Wout / 512): 8 waves x 64 positions.
    constexpr int g1 = (32  / (16 * 2)) * (32 * 176 * 608 / 512);  // 6688
    constexpr int g2 = (64  / (16 * 4)) * (32 * 88  * 304 / 512);  // 1672
    constexpr int g3 = (128 / (16 * 4)) * (32 * 44  * 152 / 512);  // 836

    conv3x3s2_wmma<4, 352, 1216, 32, 176, 608, 2, 36, false>
        <<<g1, 256, 0, stream>>>(x,  wc1, bc1, h1);
    conv3x3s2_wmma<32, 176, 608, 64, 88, 304, 4, 144, false>
        <<<g2, 256, 0, stream>>>(h1, wc2, bc2, h2);

    zero_kernel<<<(32 * 128 + 255) / 256, 256, 0, stream>>>(pooled, 32 * 128);

    conv3x3s2_wmma<64, 88, 304, 128, 44, 152, 4, 144, true>
        <<<g3, 256, 0, stream>>>(h2, wc3, bc3, pooled);   // fused conv3 + GAP

    fc_kernel<<<1, 256, 0, stream>>>(pooled, wl1, bl1, wl2, bl2, rbuf);
    svd_so3_kernel<<<1, 32, 0, stream>>>(rbuf, outp);
}